// SO3onS2_1906965479994
// MI455X (gfx1250) — compile-verified
//
#include <hip/hip_runtime.h>
#include <math.h>
#include <stdint.h>

// CDNA5 / gfx1250, wave32 only.
typedef __attribute__((ext_vector_type(2))) float v2f;   // A/B frag for V_WMMA_F32_16X16X4_F32
typedef __attribute__((ext_vector_type(8))) float v8f;   // 16x16 f32 C/D accumulator

#define LMAX   10
#define NPAIR  66
#define BATCH  16384
#define NCDIM  (LMAX + 1)        // 11
#define NMDIM  (2 * LMAX + 1)    // 21
#define DMAX   441               // max (2l1+1)(2l2+1)
#define D4MAX  444               // DMAX padded to mult of 4; NOT mult of 64 words -> no LDS bank aliasing
#define NWAVE  4                 // waves per workgroup (one per SIMD32 of the WGP)

// One workgroup (4 waves) owns one (pair, 16-batch-row tile). f and TP are
// staged once in LDS and shared; each wave owns interleaved 16-column strips
// of the GEMM  transformed(16 x d) = TP(16 x d) @ CG(d x d)  with a private
// scg region filled by ASYNC global->LDS copies (transpose done by the per-lane
// scatter). Complex TP x real CG = two fp32 WMMA chains sharing each B fragment.
__global__ __launch_bounds__(32 * NWAVE)
void so3_tp_cg_kernel(const float* __restrict__ cre,
                      const float* __restrict__ cim,
                      const float* __restrict__ cg,
                      float* __restrict__ out)
{
    const int pair = blockIdx.y;
    const int tile = blockIdx.x;            // 16-row batch tile
    const int tid  = threadIdx.x;           // 0..127
    const int lane = tid & 31;
    const int wid  = tid >> 5;              // wave id 0..3

    // ---- scalar pass: pair geometry + offsets (uniform, 66 iterations) ----
    int l1 = 0, l2 = 0, cgoff = 0, colbase = 0, ncol_total = 0;
    {
        int p = 0, off = 0, cb = 0;
        for (int a = 0; a <= LMAX; ++a) {
            for (int b = a; b <= LMAX; ++b) {
                int dd  = (2 * a + 1) * (2 * b + 1);
                int lmn = b - a;
                int lmx = (a + b < LMAX) ? (a + b) : LMAX;
                if (p == pair) { l1 = a; l2 = b; cgoff = off; colbase = cb; }
                off += dd * dd;
                cb  += lmx - lmn + 1;
                ++p;
            }
        }
        ncol_total = cb;                    // 381 for LMAX=10
    }
    const int n2    = 2 * l2 + 1;
    const int d     = (2 * l1 + 1) * n2;
    const int d4    = (d + 3) & ~3;
    const int lmin  = l2 - l1;
    const int lmaxp = (l1 + l2 < LMAX) ? (l1 + l2) : LMAX;
    const int nl    = lmaxp - lmin + 1;
    const int valid = (lmaxp + 1) * (lmaxp + 1) - lmin * lmin;
    const int b0    = tile * 16;

    // ---- LDS (~203 KB of the 320 KB WGP pool) ----
    __shared__ float sfre[16][NCDIM][NMDIM];          // 14784 B  staged f (re)
    __shared__ float sfim[16][NCDIM][NMDIM];          // 14784 B  staged f (im)
    __shared__ float stp[2][16][D4MAX];               // 56832 B  padded TP (re,im), shared by all waves
    __shared__ float scg[NWAVE][16][D4MAX];           // 113664 B per-wave transposed padded CG strip
    __shared__ float sacc[16][NCDIM][2];              //  1408 B  output accumulators
    __shared__ int   sjmap[D4MAX];                    //  1776 B  column -> packed (l, m')

    // Stage f with conjugate extension: f[l][l+m]=c[l][m]; f[l][l-m]=(-1)^m conj(c[l][m])
    for (int idx = tid; idx < 16 * NCDIM * NMDIM; idx += 32 * NWAVE) {
        int m = idx % NMDIM;
        int t = idx / NMDIM;
        int l = t % NCDIM;
        int r = t / NCDIM;
        float re = 0.f, im = 0.f;
        if (m <= 2 * l) {
            int b = b0 + r;
            int j = m - l;
            if (j >= 0) {
                re = cre[(b * NCDIM + l) * NCDIM + j];
                im = cim[(b * NCDIM + l) * NCDIM + j];
            } else {
                int   mm = -j;
                float s  = (mm & 1) ? -1.f : 1.f;
                re =  s * cre[(b * NCDIM + l) * NCDIM + mm];
                im = -s * cim[(b * NCDIM + l) * NCDIM + mm];
            }
        }
        sfre[r][l][m] = re;
        sfim[r][l][m] = im;
    }
    for (int idx = tid; idx < 16 * NCDIM * 2; idx += 32 * NWAVE)
        (&sacc[0][0][0])[idx] = 0.f;

    // Column -> (l, m') lookup (one-time; keeps sqrt/loops out of the epilogue).
    for (int j = tid; j < d4; j += 32 * NWAVE) {
        int pk = 0;
        if (j < valid) {
            int jj = j + lmin * lmin;
            int lb = (int)sqrtf((float)jj);
            while (lb * lb > jj) --lb;
            while ((lb + 1) * (lb + 1) <= jj) ++lb;
            pk = (lb << 5) | (jj - lb * lb);      // lb<=10, m'<=20
        }
        sjmap[j] = pk;
    }
    __syncthreads();

    // Build zero-padded TP once (shared): stp[c][r][k] with k = i1*n2 + i2.
    for (int idx = tid; idx < 16 * d4; idx += 32 * NWAVE) {
        int r = idx & 15;
        int k = idx >> 4;
        float tre = 0.f, tim = 0.f;
        if (k < d) {
            int i1 = (int)((unsigned)k / (unsigned)n2);
            int i2 = k - i1 * n2;
            float f1r = sfre[r][l1][i1], f1i = sfim[r][l1][i1];
            float f2r = sfre[r][l2][i2], f2i = sfim[r][l2][i2];
            tre = f1r * f2r - f1i * f2i;
            tim = f1r * f2i + f1i * f2r;
        }
        stp[0][r][k] = tre;
        stp[1][r][k] = tim;
    }

    // One-time zero of this wave's scg region: covers the k >= d pad rows,
    // which are never touched by the async copies below.
    for (int idx = lane; idx < 16 * D4MAX; idx += 32)
        (&scg[wid][0][0])[idx] = 0.f;
    __syncthreads();

    // Fragment geometry (V_WMMA_F32_16X16X4_F32):
    //   A (16x4): lane -> M = lane%16, VGPR v -> K = k0 + v + 2*(lane/16)
    //   B (4x16): lane -> N = lane%16, VGPR v -> K = k0 + v + 2*(lane/16)
    //   D (16x16): lane -> N = lane%16, VGPR v -> M = v + 8*(lane/16)
    const int rA = lane & 15;
    const int hi = lane >> 4;
    const float* __restrict__ cgp = cg + cgoff;

    // Interleaved strip ownership: wave w handles n0 = (w + i*NWAVE)*16.
    for (int n0 = wid * 16; n0 < d; n0 += 16 * NWAVE) {
        // Stage this wave's transposed CG strip with ASYNC global->LDS copies:
        //   LDS scg[wid][c][k]  <-  cg[k][n0+c]   (per-lane scatter = transpose)
        // Tracked on ASYNCcnt; overlaps with the prefetch issue below.
        for (int idx = lane; idx < 16 * d4; idx += 32) {
            int c = idx & 15;
            int k = idx >> 4;
            int j = n0 + c;
            if (k < d) {
                if (j < d) {
                    unsigned ldsa = (unsigned)(uintptr_t)&scg[wid][c][k];
                    unsigned goff = (unsigned)((k * d + j) * 4);
                    asm volatile("global_load_async_to_lds_b32 %0, %1, %2"
                                 :
                                 : "v"(ldsa), "v"(goff), "s"(cgp)
                                 : "memory");
                } else {
                    scg[wid][c][k] = 0.f;   // final-strip column overflow (stale otherwise)
                }
            }
        }

        // Warm L2->WGP path for this wave's NEXT strip while the async copies fly
        // (lowers to global_prefetch_b8; one line per CG row).
        {
            const int nn = n0 + 16 * NWAVE;
            if (nn < d) {
                for (int k = lane; k < d; k += 32)
                    __builtin_prefetch(&cgp[k * d + nn], 0, 0);
            }
        }

        // Drain the async copies before the LDS-fed WMMA chain.
        asm volatile("s_wait_asynccnt 0x0" ::: "memory");

        v8f acc_re = {};
        v8f acc_im = {};

        // Hot loop: 3 ds_load_b64 + 2 v_wmma, no guards, no division.
        for (int k0 = 0; k0 < d; k0 += 4) {
            const int kb = k0 + 2 * hi;
            v2f a_re = *(const v2f*)&stp[0][rA][kb];
            v2f a_im = *(const v2f*)&stp[1][rA][kb];
            v2f bf   = *(const v2f*)&scg[wid][rA][kb];
            acc_re = __builtin_amdgcn_wmma_f32_16x16x4_f32(
                false, a_re, false, bf, (short)0, acc_re, false, false);
            acc_im = __builtin_amdgcn_wmma_f32_16x16x4_f32(
                false, a_im, false, bf, (short)0, acc_im, false, false);
        }

        // Epilogue: out[b,l] += transformed[b,j] * conj(f[l][m'])  (table lookup)
        const int jcol = n0 + (lane & 15);
        if (jcol < valid) {
            const int pk = sjmap[jcol];
            const int lb = pk >> 5;
            const int mp = pk & 31;
            const int lr = lb - lmin;
#pragma unroll
            for (int v = 0; v < 8; ++v) {
                const int   br  = v + 8 * hi;
                const float flr = sfre[br][lb][mp];
                const float fli = sfim[br][lb][mp];
                const float tre = acc_re[v];
                const float tim = acc_im[v];
                atomicAdd(&sacc[br][lr][0], tre * flr + tim * fli);
                atomicAdd(&sacc[br][lr][1], tim * flr - tre * fli);
            }
        }
        // no barrier: scg region is private to this wave; same-wave LDS is in-order
    }

    __syncthreads();

    // Write interleaved complex output (B, ncol_total)
    for (int idx = tid; idx < 16 * nl; idx += 32 * NWAVE) {
        const int br = idx / nl;
        const int li = idx % nl;
        const size_t b = (size_t)(b0 + br);
        const size_t c = (size_t)(colbase + li);
        out[(b * (size_t)ncol_total + c) * 2 + 0] = sacc[br][li][0];
        out[(b * (size_t)ncol_total + c) * 2 + 1] = sacc[br][li][1];
    }
}

extern "C" void kernel_launch(void* const* d_in, const int* in_sizes, int n_in,
                              void* d_out, int out_size, void* d_ws, size_t ws_size,
                              hipStream_t stream)
{
    (void)in_sizes; (void)n_in; (void)d_ws; (void)ws_size; (void)out_size;
    const float* cre = (const float*)d_in[0];   // (B, 11, 11) f32
    const float* cim = (const float*)d_in[1];   // (B, 11, 11) f32
    const float* cg  = (const float*)d_in[2];   // flat CG tables f32
    float*       out = (float*)d_out;           // (B, 381) complex64 interleaved

    dim3 grid(BATCH / 16, NPAIR);
    so3_tp_cg_kernel<<<grid, 32 * NWAVE, 0, stream>>>(cre, cim, cg, out);
}